// BandedJointEncoder_88742614270699
// MI455X (gfx1250) — compile-verified
//
#include <hip/hip_runtime.h>
#include <cmath>

typedef __attribute__((ext_vector_type(2))) float v2f;
typedef __attribute__((ext_vector_type(4))) float v4f;
typedef __attribute__((ext_vector_type(8))) float v8f;

#define B_    32
#define T_    256
#define TP_   264      // padded row stride: [0]=left halo, [1..256]=data, [257..263]=right halo
#define ZDIM  32

__device__ __forceinline__ float softplusf(float x) {
    // jax.nn.softplus: max(x,0) + log1p(exp(-|x|))
    return fmaxf(x, 0.0f) + log1pf(expf(-fabsf(x)));
}

// ---------------------------------------------------------------------------
// Copy (rows x 256) into padded layout (stride TP_), zeroing the halo.
// ---------------------------------------------------------------------------
__global__ __launch_bounds__(256) void pad_rows(
    const float* __restrict__ src, float* __restrict__ dst)
{
    const int row = blockIdx.x;
    const int t   = threadIdx.x;
    const float* s = src + (size_t)row * T_;
    float*       d = dst + (size_t)row * TP_;
    d[1 + t] = s[t];
    if (t == 0)   d[0] = 0.0f;
    if (t >= 248) d[t + 9] = 0.0f;     // indices 257..264 (264 is next row's [0])
}

// ---------------------------------------------------------------------------
// W[co][ci][k]  ->  Wt[k][ci][co]   (write-coalesced; tiny, runs once)
// ---------------------------------------------------------------------------
template<int CI, int CO>
__global__ __launch_bounds__(256) void transpose_w(
    const float* __restrict__ W, float* __restrict__ Wt)
{
    const int idx = blockIdx.x * 256 + threadIdx.x;   // (k*CI + ci)*CO + co
    if (idx >= CO * CI * 3) return;
    const int co  = idx % CO;
    const int rem = idx / CO;
    const int ci  = rem % CI;
    const int k   = rem / CI;
    Wt[idx] = W[((size_t)co * CI + ci) * 3 + k];
}

// ---------------------------------------------------------------------------
// Conv1d (K=3, SAME) as WMMA f32 16x16x4 GEMM on padded rows:
//   out[b][co][1+t] = relu( sum_{ci,k} Wt[k][ci][co] * xp[b][ci][t+k] + bias[co] )
// Block = 256 threads (8 wave32). Grid = (CO/16, B).
// Each wave owns two 16-wide t-tiles; K-dim = ci in chunks of 4, 3 taps.
// All loads unconditional (halo provides the zero padding).
// ---------------------------------------------------------------------------
template<int CI, int CO, bool RELU>
__global__ __launch_bounds__(256) void conv3_wmma(
    const float* __restrict__ xp, const float* __restrict__ Wt,
    const float* __restrict__ bias, float* __restrict__ outp)
{
    const int co0   = blockIdx.x * 16;
    const int b     = blockIdx.y;
    const int lane  = threadIdx.x & 31;   // wave32
    const int wave  = threadIdx.x >> 5;
    const int mrow  = lane & 15;          // A: M row / B: N col
    const int khalf = lane >> 4;          // which K pair this lane holds
    const int t0a   = (wave * 2) * 16;
    const int t0b   = t0a + 16;

    const float* xb = xp + (size_t)b * CI * TP_;

    v8f acc0 = {};
    v8f acc1 = {};

    for (int ci0 = 0; ci0 < CI; ci0 += 4) {
        const int ciA = ci0 + 2 * khalf;       // this lane's two K (ci) rows
#pragma unroll
        for (int k = 0; k < 3; ++k) {
            // A fragment: Wt[k][ciA + {0,1}][co0+mrow]  — lane-consecutive
            const float* wt = Wt + ((size_t)(k * CI + ciA)) * CO + co0 + mrow;
            v2f afr;
            afr.x = wt[0];
            afr.y = wt[CO];

            // B fragments: xp rows ciA, ciA+1 at padded index t + k
            const float* xr0 = xb + (size_t)ciA * TP_ + k;
            const float* xr1 = xr0 + TP_;
            v2f bfa, bfb;
            bfa.x = xr0[t0a + mrow];
            bfa.y = xr1[t0a + mrow];
            bfb.x = xr0[t0b + mrow];
            bfb.y = xr1[t0b + mrow];

            acc0 = __builtin_amdgcn_wmma_f32_16x16x4_f32(
                false, afr, false, bfa, (short)0, acc0, false, false);
            acc1 = __builtin_amdgcn_wmma_f32_16x16x4_f32(
                false, afr, false, bfb, (short)0, acc1, false, false);
        }
    }

    // D layout: reg r, lanes 0-15 -> M=r, lanes 16-31 -> M=8+r; N = lane&15
    const int drow = (lane >> 4) * 8;
    const int dcol = lane & 15;
#pragma unroll
    for (int r = 0; r < 8; ++r) {
        const int co = co0 + drow + r;
        float v0 = acc0[r] + bias[co];
        float v1 = acc1[r] + bias[co];
        if (RELU) { v0 = fmaxf(v0, 0.0f); v1 = fmaxf(v1, 0.0f); }
        float* orow = outp + ((size_t)b * CO + co) * TP_ + 1;
        orow[t0a + dcol] = v0;
        orow[t0b + dcol] = v1;
    }

    // Zero the halo of this block's 16 output rows (read by the next conv).
    if (threadIdx.x < 16) {
        float* orow = outp + ((size_t)b * CO + co0 + threadIdx.x) * TP_;
        orow[0] = 0.0f;
#pragma unroll
        for (int j = 257; j < TP_; ++j) orow[j] = 0.0f;
    }
}

// ---------------------------------------------------------------------------
// Bidiagonal inverse-transpose fill + mu copy (stats in padded layout).
// scale_tril[a][c] = (1/d_a) * prod_{k=c}^{a-1} (-s_k/d_k), c <= a, else 0.
// Row recurrence: row_a = (-s_{a-1}/d_a) * row_{a-1}; L[a][a] = 1/d_a.
// One block per (b,z); 64 threads x 4 columns; coalesced b128 NT stores.
// ---------------------------------------------------------------------------
__global__ __launch_bounds__(64) void bidiag_inv_fill(
    const float* __restrict__ statsp, float* __restrict__ mu,
    float* __restrict__ L)
{
    const int bz = blockIdx.x;         // b*32 + z
    const int b  = bz >> 5;
    const int z  = bz & 31;
    const float* mu_row = statsp + ((size_t)b * 96 + z) * TP_ + 1;
    const float* d_row  = statsp + ((size_t)b * 96 + 32 + 2 * z) * TP_ + 1;
    const float* s_row  = statsp + ((size_t)b * 96 + 33 + 2 * z) * TP_ + 1;

    __shared__ float inv_d[T_];
    __shared__ float mlt[T_];

    const int tid = threadIdx.x;
    float* muo = mu + (size_t)bz * T_;
    for (int i = tid; i < T_; i += 64) {
        const float dv = softplusf(d_row[i]) + 1.0f;
        inv_d[i] = 1.0f / dv;
        const float sp = (i >= 1) ? softplusf(s_row[i - 1]) : 0.0f;
        mlt[i] = -sp / dv;
        muo[i] = mu_row[i];
    }
    __syncthreads();

    const int c0 = tid * 4;
    v4f v = {0.0f, 0.0f, 0.0f, 0.0f};
    float* Lb = L + (size_t)bz * (T_ * T_) + c0;

    for (int a = 0; a < T_; ++a) {
        const float mm = mlt[a];
        v.x *= mm; v.y *= mm; v.z *= mm; v.w *= mm;
        if (a >= c0 && a < c0 + 4) {
            const float iv = inv_d[a];
            switch (a - c0) {
                case 0: v.x = iv; break;
                case 1: v.y = iv; break;
                case 2: v.z = iv; break;
                default: v.w = iv; break;
            }
        }
        v4f w;
        w.x = __builtin_isfinite(v.x) ? v.x : 0.0f;
        w.y = __builtin_isfinite(v.y) ? v.y : 0.0f;
        w.z = __builtin_isfinite(v.z) ? v.z : 0.0f;
        w.w = __builtin_isfinite(v.w) ? v.w : 0.0f;
        // 268 MB streamed once, never re-read: bypass L2 residency
        __builtin_nontemporal_store(w, (v4f*)(Lb + (size_t)a * T_));
    }
}

// ---------------------------------------------------------------------------
extern "C" void kernel_launch(void* const* d_in, const int* in_sizes, int n_in,
                              void* d_out, int out_size, void* d_ws, size_t ws_size,
                              hipStream_t stream) {
    const float* x  = (const float*)d_in[0];
    const float* W0 = (const float*)d_in[1];
    const float* b0 = (const float*)d_in[2];
    const float* W1 = (const float*)d_in[3];
    const float* b1 = (const float*)d_in[4];
    const float* W2 = (const float*)d_in[5];
    const float* b2 = (const float*)d_in[6];

    float* ws     = (float*)d_ws;
    float* xpad   = ws;                                   // 32* 64*264
    float* h0p    = xpad  + (size_t)B_ *  64 * TP_;       // 32*256*264
    float* h1p    = h0p   + (size_t)B_ * 256 * TP_;       // 32*256*264
    float* statsp = h1p   + (size_t)B_ * 256 * TP_;       // 32* 96*264
    float* Wt0    = statsp + (size_t)B_ * 96 * TP_;       // 3* 64*256
    float* Wt1    = Wt0   + 3 *  64 * 256;                // 3*256*256
    float* Wt2    = Wt1   + 3 * 256 * 256;                // 3*256* 96

    float* out   = (float*)d_out;
    float* mu    = out;                                   // 32*32*256
    float* Ltril = out + (size_t)B_ * ZDIM * T_;          // 32*32*256*256

    // Stage inputs: padded x, transposed weights
    pad_rows<<<dim3(B_ * 64), dim3(256), 0, stream>>>(x, xpad);
    transpose_w< 64, 256><<<dim3(( 64 * 256 * 3 + 255) / 256), dim3(256), 0, stream>>>(W0, Wt0);
    transpose_w<256, 256><<<dim3((256 * 256 * 3 + 255) / 256), dim3(256), 0, stream>>>(W1, Wt1);
    transpose_w<256,  96><<<dim3((256 *  96 * 3 + 255) / 256), dim3(256), 0, stream>>>(W2, Wt2);

    dim3 blk(256);
    conv3_wmma< 64, 256, true ><<<dim3(16, B_), blk, 0, stream>>>(xpad, Wt0, b0, h0p);
    conv3_wmma<256, 256, true ><<<dim3(16, B_), blk, 0, stream>>>(h0p,  Wt1, b1, h1p);
    conv3_wmma<256,  96, false><<<dim3( 6, B_), blk, 0, stream>>>(h1p,  Wt2, b2, statsp);

    bidiag_inv_fill<<<dim3(B_ * ZDIM), dim3(64), 0, stream>>>(statsp, mu, Ltril);
}